// DGCNN_28956669510282
// MI455X (gfx1250) — compile-verified
//
#include <hip/hip_runtime.h>
#include <hip/hip_bf16.h>

// ---------------------------------------------------------------------------
// DGCNN fused implementation for gfx1250 (MI455X).
//   K1 zero counts -> K2 edge count -> K3 block scan -> K4 scan block sums
//   -> K5 finalize offsets -> K6 scatter CSR -> K7 fused per-graph DGCNN
// K7: one workgroup (256 thr = 8 wave32) per graph.
//   * GCN matmuls: V_WMMA_F32_16X16X4_F32 on LDS-resident 128x64 tiles
//   * Weight staging: TENSOR_LOAD_TO_LDS (TDM) with hardware LDS padding,
//     issued by wave 0 and overlapped with edge aggregation (TENSORcnt).
// ---------------------------------------------------------------------------

#define NNODES   131072
#define NEDGES   2097152
#define NGRAPHS  1024
#define NPG      128
#define HID      64
#define KSORT    30
#define DLAT     193            // 3*H + 1
#define DENSE    352            // 32 * 11

typedef float v2f __attribute__((ext_vector_type(2)));
typedef float v8f __attribute__((ext_vector_type(8)));
typedef unsigned int u32x4 __attribute__((ext_vector_type(4)));
typedef int i32x8 __attribute__((ext_vector_type(8)));
typedef int i32x4 __attribute__((ext_vector_type(4)));

struct __align__(16) f4 { float x, y, z, w; };

// ---------------- preprocessing kernels (CSR build, grouped by dst) --------

__global__ void k_zero(int* p, int n) {
    int i = blockIdx.x * blockDim.x + threadIdx.x;
    if (i < n) p[i] = 0;
}

__global__ void k_count(const int* __restrict__ ei, int* __restrict__ cnt, int E) {
    int e = blockIdx.x * blockDim.x + threadIdx.x;
    if (e < E) atomicAdd(&cnt[ei[E + e]], 1);   // dst row of edge_index
}

__global__ void k_scan_block(const int* __restrict__ cnt, int* __restrict__ off,
                             int* __restrict__ bsum) {
    __shared__ int sd[256];
    int t = threadIdx.x;
    int i = blockIdx.x * 256 + t;
    int v = cnt[i];
    sd[t] = v;
    __syncthreads();
    for (int ofs = 1; ofs < 256; ofs <<= 1) {
        int tmp = (t >= ofs) ? sd[t - ofs] : 0;
        __syncthreads();
        sd[t] += tmp;
        __syncthreads();
    }
    off[i] = sd[t] - v;                 // exclusive within block
    if (t == 255) bsum[blockIdx.x] = sd[255];
}

__global__ void k_scan_bsum(int* __restrict__ bsum, int n) {
    __shared__ int sd[512];
    int t = threadIdx.x;
    int v = (t < n) ? bsum[t] : 0;
    sd[t] = v;
    __syncthreads();
    for (int ofs = 1; ofs < 512; ofs <<= 1) {
        int tmp = (t >= ofs) ? sd[t - ofs] : 0;
        __syncthreads();
        sd[t] += tmp;
        __syncthreads();
    }
    if (t < n) bsum[t] = sd[t] - v;     // exclusive
}

__global__ void k_finalize(int* __restrict__ off, const int* __restrict__ bsum,
                           int* __restrict__ cnt, int n, int etot) {
    int i = blockIdx.x * blockDim.x + threadIdx.x;
    if (i < n) {
        off[i] += bsum[i >> 8];
        cnt[i] = 0;                     // reused as scatter cursor
        if (i == n - 1) off[n] = etot;
    }
}

__global__ void k_scatter(const int* __restrict__ ei, const int* __restrict__ off,
                          int* __restrict__ cnt, int* __restrict__ csr, int E) {
    int e = blockIdx.x * blockDim.x + threadIdx.x;
    if (e < E) {
        int d = ei[E + e];
        int s = ei[e];
        int pos = off[d] + atomicAdd(&cnt[d], 1);
        csr[pos] = s;
    }
}

// ---------------- TDM helpers ----------------------------------------------
// Build a Tensor DMA Descriptor (D#) per CDNA5 ISA §8.3-8.6 and issue
// TENSOR_LOAD_TO_LDS.  2-D fp32 tile, optional "pad 1 DWORD every 64 DWORDs"
// (pad_interval=5, pad_amount=0) to produce a 65-float LDS row stride.
// Caller must be a single wave; completion via s_wait_tensorcnt.
// (clang-23 / therock-10.0 6-arg builtin form: groups 2/3 are NULL for 2-D
//  tensors, extra i32x8 group zero-filled.)

__device__ __forceinline__ void tdm_load(const void* gptr, unsigned lds_byte,
                                         unsigned dim0, unsigned dim1,
                                         unsigned tile0, unsigned tile1,
                                         unsigned stride0, bool pad) {
    unsigned long long ga = (unsigned long long)gptr;
    u32x4 g0;
    g0[0] = 1u;                                   // count=1, user mode
    g0[1] = lds_byte;                             // lds_addr
    g0[2] = (unsigned)(ga & 0xffffffffu);         // global_addr[31:0]
    g0[3] = (unsigned)((ga >> 32) & 0x01ffffffu)  // global_addr[56:32]
          | (2u << 30);                           // type = 2 (image)
    i32x8 g1;
    g1[0] = (int)((2u << 16) |                    // data_size = 4B
                  (pad ? ((1u << 20) | (5u << 22)) : 0u)); // pad_en, ivl=64dw, amt=1dw
    g1[1] = (int)((dim0 & 0xffffu) << 16);        // tensor_dim0[15:0]
    g1[2] = (int)(((dim0 >> 16) & 0xffffu) | ((dim1 & 0xffffu) << 16));
    g1[3] = (int)(((dim1 >> 16) & 0xffffu) | ((tile0 & 0xffffu) << 16));
    g1[4] = (int)(tile1 & 0xffffu);               // tile_dim1 (tile_dim2=0)
    g1[5] = (int)stride0;                         // tensor_dim0_stride[31:0]
    g1[6] = 0;
    g1[7] = 0;
    i32x4 zz4 = {0, 0, 0, 0};
    i32x8 zz8 = {0, 0, 0, 0, 0, 0, 0, 0};
    __builtin_amdgcn_tensor_load_to_lds(g0, g1, zz4, zz4, zz8, 0);
}

// ---------------- fused per-graph DGCNN kernel -----------------------------

// LDS layout (floats):
#define XC_OFF   0                       // 128 x 193 concat/feature buffer
#define XC_SZ    (128 * 193)             // 24704
#define HB_OFF   (XC_OFF + XC_SZ)        // 128 x 68 matmul output (padded)
#define HB_STR   68
#define HB_SZ    (128 * HB_STR)          // 8704
#define WB_OFF   (HB_OFF + HB_SZ)        // 64 x 65 weight staging (reused)
#define WB_STR   65
#define WB_SZ    (64 * WB_STR)           // 4160
#define DI_OFF   (WB_OFF + WB_SZ)        // dinv[128]
#define H3_OFF   (DI_OFF + 128)          // h3[128] / lin2 reduction
#define SM_TOT   (H3_OFF + 128)          // 37824 floats = 151,296 B

// conv-head scratch overlaid on HB region (GCN h-buffer dead by then)
#define C_H1     (HB_OFF)                // 16 x 30
#define C_P      (HB_OFF + 512)          // 16 x 15
#define C_FLAT   (HB_OFF + 1024)         // 352
#define C_HM     (HB_OFF + 1408)         // 128

__launch_bounds__(256)
__global__ void k_dgcnn(const int* __restrict__ z,
                        const int* __restrict__ off,
                        const int* __restrict__ csr,
                        const float* __restrict__ ztab,
                        const float* __restrict__ W0, const float* __restrict__ b0,
                        const float* __restrict__ W1, const float* __restrict__ b1,
                        const float* __restrict__ W2, const float* __restrict__ b2,
                        const float* __restrict__ W3, const float* __restrict__ b3,
                        const float* __restrict__ c1w, const float* __restrict__ c1b,
                        const float* __restrict__ c2w, const float* __restrict__ c2b,
                        const float* __restrict__ l1w, const float* __restrict__ l1b,
                        const float* __restrict__ l2w, const float* __restrict__ l2b,
                        float* __restrict__ out) {
    __shared__ float sm[SM_TOT];
    __shared__ int sel[KSORT];

    const int g     = blockIdx.x;
    const int tid   = threadIdx.x;
    const int lane  = tid & 31;
    const int wv    = tid >> 5;
    const int half  = lane >> 4;
    const int lm    = lane & 15;
    const int gbase = g * NPG;

    float* XC = &sm[XC_OFF];
    float* HB = &sm[HB_OFF];
    float* WB = &sm[WB_OFF];
    float* DI = &sm[DI_OFF];
    float* H3 = &sm[H3_OFF];
    const unsigned wb_lds = (unsigned)(unsigned long long)(&sm[WB_OFF]);

    // ---- TDM: prefetch W0 into padded LDS (wave 0 only; EXEC ignored) ----
    if (wv == 0) {
        tdm_load(W0, wb_lds, HID, HID, HID, HID, HID, true);
    }

    // ---- embedding gather: z_table rows -> XC columns 0:64 ----
    {
        const int v  = tid >> 1;
        const int cb = (tid & 1) * 32;
        const int zi = z[gbase + v];
        const f4* src = (const f4*)&ztab[zi * HID + cb];
        float* dst = &XC[v * DLAT + cb];
        #pragma unroll
        for (int r = 0; r < 8; r++) {
            f4 t = src[r];
            dst[r * 4 + 0] = t.x; dst[r * 4 + 1] = t.y;
            dst[r * 4 + 2] = t.z; dst[r * 4 + 3] = t.w;
        }
    }
    // ---- symmetric-norm degree factors ----
    if (tid < NPG) {
        int n   = gbase + tid;
        int deg = off[n + 1] - off[n] + 1;     // +1 self-loop
        DI[tid] = rsqrtf((float)deg);
    }

    // ---- three 64->64 GCN layers (WMMA fp32) ----
    for (int L = 0; L < 3; L++) {
        const float* bg = (L == 0) ? b0 : (L == 1) ? b1 : b2;
        if (wv == 0) __builtin_amdgcn_s_wait_tensorcnt(0);  // W_L in LDS
        __syncthreads();

        const int acol = (L == 0) ? 0 : (L - 1) * HID;  // input column block
        // 32 output tiles (8 M-tiles x 4 N-tiles), 4 per wave
        for (int t = wv; t < 32; t += 8) {
            const int mt = t >> 2, nt = t & 3;
            const int ncol = nt * 16 + lm;
            const float* Ap = &XC[(mt * 16 + lm) * DLAT + acol + 2 * half];
            v8f c = {};
            #pragma unroll
            for (int kk = 0; kk < 16; kk++) {
                const int k0 = kk * 4;
                v2f a = {Ap[k0], Ap[k0 + 1]};
                v2f b = {WB[(k0 + 2 * half) * WB_STR + ncol],
                         WB[(k0 + 2 * half + 1) * WB_STR + ncol]};
                c = __builtin_amdgcn_wmma_f32_16x16x4_f32(
                        false, a, false, b, (short)0, c, false, false);
            }
            #pragma unroll
            for (int r = 0; r < 8; r++)
                HB[(mt * 16 + half * 8 + r) * HB_STR + ncol] = c[r];
        }
        __syncthreads();

        // WB is dead now: wave 0 kicks off the next weight DMA, overlapped
        // with the aggregation phase below.
        if (wv == 0) {
            if (L == 0)      tdm_load(W1, wb_lds, HID, HID, HID, HID, HID, true);
            else if (L == 1) tdm_load(W2, wb_lds, HID, HID, HID, HID, HID, true);
            else             tdm_load(c1w, wb_lds, 16 * DLAT, 1, 16 * DLAT, 1,
                                      16 * DLAT, false);   // conv1 weights, flat
        }

        // ---- aggregation: x[v] = tanh(b + sum_e norm*h[src] + dinv^2*h[v]) ----
        {
            const int v  = tid >> 1;
            const int cb = (tid & 1) * 32;
            const int n  = gbase + v;
            const int e0 = off[n], e1 = off[n + 1];
            const float dv = DI[v];
            const float sn = dv * dv;
            f4 acc[8];
            const f4* hv = (const f4*)&HB[v * HB_STR + cb];
            #pragma unroll
            for (int r = 0; r < 8; r++) {
                f4 h = hv[r];
                acc[r].x = sn * h.x; acc[r].y = sn * h.y;
                acc[r].z = sn * h.z; acc[r].w = sn * h.w;
            }
            for (int e = e0; e < e1; e++) {
                const int s = csr[e] & (NPG - 1);
                const float w = DI[s] * dv;
                const f4* hs = (const f4*)&HB[s * HB_STR + cb];
                #pragma unroll
                for (int r = 0; r < 8; r++) {
                    f4 h = hs[r];
                    acc[r].x += w * h.x; acc[r].y += w * h.y;
                    acc[r].z += w * h.z; acc[r].w += w * h.w;
                }
            }
            float* dst = &XC[v * DLAT + L * HID + cb];
            #pragma unroll
            for (int r = 0; r < 8; r++) {
                dst[r * 4 + 0] = tanhf(acc[r].x + bg[cb + r * 4 + 0]);
                dst[r * 4 + 1] = tanhf(acc[r].y + bg[cb + r * 4 + 1]);
                dst[r * 4 + 2] = tanhf(acc[r].z + bg[cb + r * 4 + 2]);
                dst[r * 4 + 3] = tanhf(acc[r].w + bg[cb + r * 4 + 3]);
            }
        }
        __syncthreads();
    }

    // ---- layer 3: 64 -> 1 ----
    if (tid < NPG) {
        float a = 0.f;
        #pragma unroll 8
        for (int k = 0; k < HID; k++) a += XC[tid * DLAT + 128 + k] * W3[k];
        H3[tid] = a;
    }
    __syncthreads();
    if (tid < NPG) {
        const int n = gbase + tid;
        const float dv = DI[tid];
        float a = dv * dv * H3[tid];
        for (int e = off[n]; e < off[n + 1]; e++) {
            const int s = csr[e] & (NPG - 1);
            a += DI[s] * dv * H3[s];
        }
        XC[tid * DLAT + 192] = tanhf(a + b3[0]);
    }
    __syncthreads();

    // ---- global_sort_pool: stable descending rank on last channel ----
    if (tid < NPG) {
        const float kv = XC[tid * DLAT + 192];
        int rank = 0;
        for (int u = 0; u < NPG; u++) {
            const float ku = XC[u * DLAT + 192];
            rank += (ku > kv) || (ku == kv && u < tid);
        }
        if (rank < KSORT) sel[rank] = tid;
    }
    __syncthreads();

    // ---- conv1 (per-node linear D->16) + ReLU; c1w already DMAed to WB ----
    if (wv == 0) __builtin_amdgcn_s_wait_tensorcnt(0);
    __syncthreads();
    for (int o = tid; o < 16 * KSORT; o += 256) {
        const int c = o / KSORT, kk = o % KSORT;
        const int nd = sel[kk];
        float a = c1b[c];
        for (int d = 0; d < DLAT; d++) a += XC[nd * DLAT + d] * WB[c * DLAT + d];
        sm[C_H1 + c * KSORT + kk] = fmaxf(a, 0.f);
    }
    __syncthreads();
    // conv2 weights DMA overlapped with maxpool
    if (wv == 0) tdm_load(c2w, wb_lds, 32 * 16 * 5, 1, 32 * 16 * 5, 1,
                          32 * 16 * 5, false);
    // ---- maxpool1d(2,2): 30 -> 15 ----
    for (int o = tid; o < 16 * 15; o += 256) {
        const int c = o / 15, j = o % 15;
        sm[C_P + c * 15 + j] = fmaxf(sm[C_H1 + c * KSORT + 2 * j],
                                     sm[C_H1 + c * KSORT + 2 * j + 1]);
    }
    if (wv == 0) __builtin_amdgcn_s_wait_tensorcnt(0);
    __syncthreads();
    // ---- conv2: 16 -> 32, k=5 -> length 11, ReLU ----
    for (int o = tid; o < 32 * 11; o += 256) {
        const int co = o / 11, t2 = o % 11;
        float a = c2b[co];
        #pragma unroll
        for (int ci = 0; ci < 16; ci++)
            #pragma unroll
            for (int q = 0; q < 5; q++)
                a += sm[C_P + ci * 15 + t2 + q] * WB[(co * 16 + ci) * 5 + q];
        sm[C_FLAT + co * 11 + t2] = fmaxf(a, 0.f);   // channel-major flatten
    }
    __syncthreads();
    // ---- lin1: 352 -> 128, ReLU ----
    if (tid < 128) {
        float a = l1b[tid];
        for (int i = 0; i < DENSE; i++) {
            if ((i & 7) == 0 && i + 8 < DENSE)
                __builtin_prefetch(&l1w[(i + 8) * 128 + tid]);   // global_prefetch_b8
            a += sm[C_FLAT + i] * l1w[i * 128 + tid];
        }
        sm[C_HM + tid] = fmaxf(a, 0.f);
    }
    __syncthreads();
    // ---- lin2: 128 -> 1 ----
    if (tid < 128) H3[tid] = sm[C_HM + tid] * l2w[tid];
    __syncthreads();
    for (int s = 64; s > 0; s >>= 1) {
        if (tid < s) H3[tid] += H3[tid + s];
        __syncthreads();
    }
    if (tid == 0) out[g] = H3[0] + l2b[0];
}

// ---------------------------------------------------------------------------

extern "C" void kernel_launch(void* const* d_in, const int* in_sizes, int n_in,
                              void* d_out, int out_size, void* d_ws, size_t ws_size,
                              hipStream_t stream) {
    const int*   z    = (const int*)d_in[0];
    const int*   ei   = (const int*)d_in[1];
    const float* ztab = (const float*)d_in[3];
    const float* W0   = (const float*)d_in[4];
    const float* b0   = (const float*)d_in[5];
    const float* W1   = (const float*)d_in[6];
    const float* b1   = (const float*)d_in[7];
    const float* W2   = (const float*)d_in[8];
    const float* b2   = (const float*)d_in[9];
    const float* W3   = (const float*)d_in[10];
    const float* b3   = (const float*)d_in[11];
    const float* c1w  = (const float*)d_in[12];
    const float* c1b  = (const float*)d_in[13];
    const float* c2w  = (const float*)d_in[14];
    const float* c2b  = (const float*)d_in[15];
    const float* l1w  = (const float*)d_in[16];
    const float* l1b  = (const float*)d_in[17];
    const float* l2w  = (const float*)d_in[18];
    const float* l2b  = (const float*)d_in[19];
    float* out = (float*)d_out;

    const int N = NNODES, E = NEDGES;

    // workspace layout
    char* ws = (char*)d_ws;
    int* cnt  = (int*)(ws + 0);                         // N ints
    int* off  = (int*)(ws + 524288);                    // N+1 ints
    int* bsum = (int*)(ws + 1048832);                   // 512 ints
    int* csr  = (int*)(ws + 1051136);                   // E ints

    k_zero<<<N / 256, 256, 0, stream>>>(cnt, N);
    k_count<<<E / 256, 256, 0, stream>>>(ei, cnt, E);
    k_scan_block<<<N / 256, 256, 0, stream>>>(cnt, off, bsum);
    k_scan_bsum<<<1, 512, 0, stream>>>(bsum, N / 256);
    k_finalize<<<N / 256, 256, 0, stream>>>(off, bsum, cnt, N, E);
    k_scatter<<<E / 256, 256, 0, stream>>>(ei, off, cnt, csr, E);

    k_dgcnn<<<NGRAPHS, 256, 0, stream>>>(z, off, csr, ztab,
                                         W0, b0, W1, b1, W2, b2, W3, b3,
                                         c1w, c1b, c2w, c2b,
                                         l1w, l1b, l2w, l2b, out);
}